// AsymmetricHausdorffDistance_89764816486830
// MI455X (gfx1250) — compile-verified
//
#include <hip/hip_runtime.h>

typedef __attribute__((ext_vector_type(2))) float v2f;
typedef __attribute__((ext_vector_type(8))) float v8f;

#define TSPLIT 4   // target-dimension split: 4x more waves for occupancy

// One prep kernel: pack B operand table tb[j] = (x,y,z,t2), init per-row min
// array to +inf bits, and zero the scalar output. Re-run every call so graph
// replays are deterministic.
__global__ __launch_bounds__(256)
void hausdorff_prep_kernel(const float4* __restrict__ targ,
                           float4* __restrict__ tb,
                           unsigned int* __restrict__ rowmin,
                           unsigned int* __restrict__ out,
                           int n_targ, int n_pred) {
  const int j = blockIdx.x * blockDim.x + threadIdx.x;
  if (j < n_targ) {
    const float4 q = targ[j];
    float4 o;
    o.x = q.x; o.y = q.y; o.z = q.z;
    o.w = q.x * q.x + q.y * q.y + q.z * q.z;
    tb[j] = o;
  }
  if (j < n_pred) rowmin[j] = 0x7f800000u;   // +inf
  if (j == 0) out[0] = 0u;                    // +0.0f
}

// One wave32 owns (pred tile, target chunk). V_WMMA_F32_16X16X4_F32 computes
//   S[m][n] = t2_n - 2 p_m . t_n     (A row = {-2x,-2y,-2z,1}, B col = {x,y,z,t2})
// per-row min of S over the chunk, +p2_m, clamp >= 0, then integer atomicMin
// into rowmin[] (bit pattern order-preserving for values >= 0).
//
// Wave32 VGPR layouts (ISA 7.12.2):
//   A 16x4 f32 : 2 VGPRs; lanes 0-15 K={0,1}, lanes 16-31 K={2,3}; M = lane%16
//   B 4x16 f32 : 2 VGPRs; lanes 0-15 K={0,1}, lanes 16-31 K={2,3}; N = lane%16
//   C 16x16 f32: 8 VGPRs; VGPR v -> row v (lanes 0-15) / v+8 (lanes 16-31)
__global__ __launch_bounds__(256)
void hausdorff_wmma_kernel(const float4* __restrict__ pred,
                           const v2f* __restrict__ tb2,
                           unsigned int* __restrict__ rowmin,
                           int pred_tiles, int targ_tiles)
{
  const int lane = threadIdx.x & 31;
  const int wave = (blockIdx.x * blockDim.x + threadIdx.x) >> 5;
  if (wave >= pred_tiles) return;          // wave-uniform: EXEC stays all-1s

  const int split = blockIdx.y;            // which target chunk
  const int tbeg = (int)(((long long)targ_tiles * split) / TSPLIT);
  const int tend = (int)(((long long)targ_tiles * (split + 1)) / TSPLIT);

  const int half = lane >> 4;
  const int l16  = lane & 15;

  // --- A fragment (branchless, once per wave) ---
  const float4 p = pred[wave * 16 + l16];
  const float p2lane = p.x * p.x + p.y * p.y + p.z * p.z;
  v2f a;
  a.x = (half == 0) ? (-2.0f * p.x) : (-2.0f * p.z);
  a.y = (half == 0) ? (-2.0f * p.y) : 1.0f;

  // lane's 8-byte B fragment of tile t sits at bp[t*32]
  const v2f* bp = tb2 + (l16 * 2 + half);

  float mins[8];
#pragma unroll
  for (int v = 0; v < 8; ++v) mins[v] = __builtin_inff();

  int t = tbeg;
  if (t + 4 <= tend) {
    // software-pipelined: preload next group while WMMAs consume current one
    v2f nb0 = bp[(t + 0) * 32];
    v2f nb1 = bp[(t + 1) * 32];
    v2f nb2 = bp[(t + 2) * 32];
    v2f nb3 = bp[(t + 3) * 32];
    for (; t + 8 <= tend; t += 4) {
      const v2f b0 = nb0, b1 = nb1, b2 = nb2, b3 = nb3;
      nb0 = bp[(t + 4) * 32];
      nb1 = bp[(t + 5) * 32];
      nb2 = bp[(t + 6) * 32];
      nb3 = bp[(t + 7) * 32];
      v8f c0 = {}, c1 = {}, c2 = {}, c3 = {};
      c0 = __builtin_amdgcn_wmma_f32_16x16x4_f32(false, a, false, b0, (short)0, c0, false, false);
      c1 = __builtin_amdgcn_wmma_f32_16x16x4_f32(false, a, false, b1, (short)0, c1, false, false);
      c2 = __builtin_amdgcn_wmma_f32_16x16x4_f32(false, a, false, b2, (short)0, c2, false, false);
      c3 = __builtin_amdgcn_wmma_f32_16x16x4_f32(false, a, false, b3, (short)0, c3, false, false);
#pragma unroll
      for (int v = 0; v < 8; ++v) {
        mins[v] = fminf(mins[v], fminf(c0[v], c1[v]));   // v_min3_num_f32
        mins[v] = fminf(mins[v], fminf(c2[v], c3[v]));
      }
    }
    {   // drain the pipelined group
      v8f c0 = {}, c1 = {}, c2 = {}, c3 = {};
      c0 = __builtin_amdgcn_wmma_f32_16x16x4_f32(false, a, false, nb0, (short)0, c0, false, false);
      c1 = __builtin_amdgcn_wmma_f32_16x16x4_f32(false, a, false, nb1, (short)0, c1, false, false);
      c2 = __builtin_amdgcn_wmma_f32_16x16x4_f32(false, a, false, nb2, (short)0, c2, false, false);
      c3 = __builtin_amdgcn_wmma_f32_16x16x4_f32(false, a, false, nb3, (short)0, c3, false, false);
#pragma unroll
      for (int v = 0; v < 8; ++v) {
        mins[v] = fminf(mins[v], fminf(c0[v], c1[v]));
        mins[v] = fminf(mins[v], fminf(c2[v], c3[v]));
      }
      t += 4;
    }
  }
  for (; t < tend; ++t) {
    const v2f b0 = bp[t * 32];
    v8f c0 = {};
    c0 = __builtin_amdgcn_wmma_f32_16x16x4_f32(false, a, false, b0, (short)0, c0, false, false);
#pragma unroll
    for (int v = 0; v < 8; ++v) mins[v] = fminf(mins[v], c0[v]);
  }

  // --- epilogue: per-row min across each 16-lane half, +p2, clamp, atomicMin ---
#pragma unroll
  for (int v = 0; v < 8; ++v) {
    float mv = mins[v];
    mv = fminf(mv, __shfl_xor(mv, 1, 16));
    mv = fminf(mv, __shfl_xor(mv, 2, 16));
    mv = fminf(mv, __shfl_xor(mv, 4, 16));
    mv = fminf(mv, __shfl_xor(mv, 8, 16));
    const float p2row = __shfl(p2lane, v + 8 * half, 16);   // p2 of row v+8*half
    const float pm = fmaxf(mv + p2row, 0.0f);               // partial min d^2, >= 0
    // lane v (half 0) owns row v; lane 16+v (half 1) owns row v+8
    if (l16 == v) {
      const int row = wave * 16 + v + 8 * half;
      atomicMin((int*)&rowmin[row], __float_as_int(pm));
    }
  }
}

// Final reduction: out = max over rows of sqrt(rowmin) (sqrt is monotone).
__global__ __launch_bounds__(256)
void hausdorff_final_kernel(const unsigned int* __restrict__ rowmin,
                            float* __restrict__ out, int n) {
  const int j = blockIdx.x * blockDim.x + threadIdx.x;
  float d = 0.0f;
  if (j < n) d = __builtin_sqrtf(__int_as_float((int)rowmin[j]));
  d = fmaxf(d, __shfl_xor(d, 1, 32));
  d = fmaxf(d, __shfl_xor(d, 2, 32));
  d = fmaxf(d, __shfl_xor(d, 4, 32));
  d = fmaxf(d, __shfl_xor(d, 8, 32));
  d = fmaxf(d, __shfl_xor(d, 16, 32));
  if ((threadIdx.x & 31) == 0)
    atomicMax((int*)out, __float_as_int(d));   // all values >= 0
}

extern "C" void kernel_launch(void* const* d_in, const int* in_sizes, int n_in,
                              void* d_out, int out_size, void* d_ws, size_t ws_size,
                              hipStream_t stream) {
  (void)n_in; (void)out_size; (void)ws_size;
  const float4* pred = (const float4*)d_in[0];
  const float4* targ = (const float4*)d_in[1];
  float* out = (float*)d_out;

  const int n_pred = in_sizes[0] / 4;        // 24000
  const int n_targ = in_sizes[1] / 4;        // 24000
  const int pred_tiles = n_pred / 16;        // 1500
  const int targ_tiles = n_targ / 16;        // 1500

  // workspace layout: [packed tb: n_targ*16B][rowmin: n_pred*4B]
  float4* tb = (float4*)d_ws;
  unsigned int* rowmin = (unsigned int*)((char*)d_ws + (size_t)n_targ * 16);

  const int nmax = n_targ > n_pred ? n_targ : n_pred;
  hausdorff_prep_kernel<<<(nmax + 255) / 256, 256, 0, stream>>>(
      targ, tb, rowmin, (unsigned int*)out, n_targ, n_pred);

  const int waves_per_block = 256 / 32;      // 8 waves per workgroup
  dim3 grid((pred_tiles + waves_per_block - 1) / waves_per_block, TSPLIT, 1);
  hausdorff_wmma_kernel<<<grid, 256, 0, stream>>>(pred, (const v2f*)tb, rowmin,
                                                  pred_tiles, targ_tiles);

  hausdorff_final_kernel<<<(n_pred + 255) / 256, 256, 0, stream>>>(rowmin, out, n_pred);
}